// Sampler_21775484191208
// MI455X (gfx1250) — compile-verified
//
#include <hip/hip_runtime.h>
#include <stdint.h>

// ---------------- configuration ----------------
#define NT      512                    // threads per block (16 waves)
#define TILE    4096                   // floats per streamed tile (16 KB)
#define ITERS   (TILE / (NT * 4))      // async b128 issues per thread per tile == 2
#define TOPN    50                     // exact top-N selected per row
#define CAND_CAP 64
#define NB      2048                   // radix histogram bins (11 bits max level)

// ---------------- LDS layout (bytes) ----------------
#define OFF_BUF0   0        // 16384
#define OFF_BUF1   16384    // 16384
#define OFF_HIST   32768    // 2048 * 4 = 8192
#define OFF_REDM   40960    // 512 * 4
#define OFF_REDS   43008
#define OFF_REDAV  45056
#define OFF_REDAI  47104
#define OFF_CANDV  49152    // 64 * 4
#define OFF_CANDI  49408
#define OFF_TIEV   49664
#define OFF_TIEI   49920
#define OFF_UV     50176
#define OFF_SCAL   50432    // 16 * 4 (+pad)
#define OFF_DUMMY  50688    // 32 lanes * 16B = 512
#define OFF_SCAN   51200    // 512 * 4 = 2048
#define SMEM_BYTES 53248    // < 64 KB -> no opt-in attribute needed

// Generic pointer to an LDS object: low 32 bits of the flat address are the
// LDS byte offset (aperture hi bits live in [63:32] on CDNA5).
__device__ __forceinline__ uint32_t lds_off(const void* p) {
    return (uint32_t)(uintptr_t)p;
}

// Monotone float -> uint32 key (larger float => larger key). No NaNs in input.
__device__ __forceinline__ uint32_t fkey(float x) {
    uint32_t b = __float_as_uint(x);
    return (b & 0x80000000u) ? ~b : (b | 0x80000000u);
}

// Double-buffered async global->LDS streaming over one row.
// Each thread consumes exactly the LDS slots it requested, so correctness only
// needs the per-wave ASYNCcnt wait (loads complete in order). Lanes past the
// tile end are redirected to a dummy LDS slot so every wave issues exactly
// ITERS async instructions per tile (constant-immediate waits).
template <typename F>
__device__ __forceinline__ void stream_row(const float* __restrict__ row, int V,
                                           char* smem, int tid, F f) {
    const uint64_t base = (uint64_t)(uintptr_t)row;
    const uint32_t b0 = lds_off(smem + OFF_BUF0);
    const uint32_t b1 = lds_off(smem + OFF_BUF1);
    const uint32_t dm = lds_off(smem + OFF_DUMMY) + ((uint32_t)(tid & 31) << 4);
    const int ntiles = (V + TILE - 1) / TILE;

    auto issue = [&](int t) {
        const int start = t * TILE;
        const int cnt = min(TILE, V - start);
        const uint32_t bb = (t & 1) ? b1 : b0;
#pragma unroll
        for (int r = 0; r < ITERS; ++r) {
            int i = tid * 4 + r * (NT * 4);
            uint32_t l = (i < cnt) ? (bb + ((uint32_t)i << 2)) : dm;
            uint32_t g = (i < cnt) ? ((uint32_t)(start + i) << 2) : 0u;
            asm volatile("global_load_async_to_lds_b128 %0, %1, %2"
                         :: "v"(l), "v"(g), "s"(base) : "memory");
        }
    };

    issue(0);
    for (int t = 0; t < ntiles; ++t) {
        if (t + 1 < ntiles) {
            issue(t + 1);
            asm volatile("s_wait_asynccnt 2" ::: "memory");  // tile t complete
        } else {
            asm volatile("s_wait_asynccnt 0" ::: "memory");
        }
        const int start = t * TILE;
        const int cnt = min(TILE, V - start);
        const float4* bp = (const float4*)(smem + ((t & 1) ? OFF_BUF1 : OFF_BUF0));
#pragma unroll
        for (int r = 0; r < ITERS; ++r) {
            int i = tid * 4 + r * (NT * 4);
            if (i < cnt) f(start + i, bp[i >> 2]);
        }
    }
}

// Block-parallel radix-level selection: find bin b such that
//   suffix(b+1) < remaining <= suffix(b)
// over 2048 bins, via per-thread 4-bin chunks + Hillis-Steele suffix scan.
// Updates scal[0] (prefix |= b << sh), scal[1] (remaining), scal[2] (countGreater).
__device__ __forceinline__ void select_bin(uint32_t* hist, uint32_t* scal,
                                           uint32_t* scan, int sh, int tid) {
    const uint32_t rem = scal[1];   // read BEFORE any write to scal (barriers follow)
    uint32_t h[4];
    uint32_t csum = 0;
#pragma unroll
    for (int i = 0; i < 4; ++i) { h[i] = hist[tid * 4 + i]; csum += h[i]; }
    scan[tid] = csum;
    __syncthreads();
    for (int off = 1; off < NT; off <<= 1) {
        uint32_t v = scan[tid];
        if (tid + off < NT) v += scan[tid + off];
        __syncthreads();
        scan[tid] = v;
        __syncthreads();
    }
    // scan[tid] = inclusive suffix over chunks >= tid
    uint32_t excl = scan[tid] - csum;   // sum over all bins in higher chunks
#pragma unroll
    for (int i = 3; i >= 0; --i) {      // high bin -> low bin within chunk
        uint32_t incl = excl + h[i];
        if (incl >= rem && excl < rem) {  // unique thread/bin satisfies this
            scal[0] |= (uint32_t)(tid * 4 + i) << sh;
            scal[1] = rem - excl;
            scal[2] += excl;
        }
        excl = incl;
    }
    __syncthreads();
}

__global__ void __launch_bounds__(NT)
sampler_kernel(const float* __restrict__ logits,
               const float* __restrict__ temperature,
               const float* __restrict__ top_p,
               const float* __restrict__ u,
               const int* __restrict__ top_k,
               const int* __restrict__ num_logprobs,
               char* __restrict__ out, int B, int V) {
    extern __shared__ char smem[];
    const int b   = blockIdx.x;
    const int tid = threadIdx.x;
    const float* row = logits + (size_t)b * (size_t)V;

    uint32_t* hist  = (uint32_t*)(smem + OFF_HIST);
    float*    redM  = (float*)(smem + OFF_REDM);
    float*    redS  = (float*)(smem + OFF_REDS);
    float*    redAV = (float*)(smem + OFF_REDAV);
    int*      redAI = (int*)(smem + OFF_REDAI);
    float*    candV = (float*)(smem + OFF_CANDV);
    int*      candI = (int*)(smem + OFF_CANDI);
    float*    tieV  = (float*)(smem + OFF_TIEV);
    int*      tieI  = (int*)(smem + OFF_TIEI);
    float*    uvals = (float*)(smem + OFF_UV);
    uint32_t* scal  = (uint32_t*)(smem + OFF_SCAL);
    uint32_t* scan  = (uint32_t*)(smem + OFF_SCAN);
    // scal: [0]=prefix [1]=remaining [2]=countGreater [3]=nG [4]=nTies [5]=total

    for (int i = tid; i < NB; i += NT) hist[i] = 0u;
    if (tid == 0) { scal[0] = 0u; scal[1] = TOPN; scal[2] = 0u; }
    __syncthreads();

    // ---- Pass 1 (HBM): online max / argmax / sum-exp + radix level-0 hist ----
    float lm = -INFINITY, lsum = 0.0f, av = -INFINITY;
    int ai = 0;
    stream_row(row, V, smem, tid, [&](int idx, float4 v) {
        float xs[4] = {v.x, v.y, v.z, v.w};
#pragma unroll
        for (int e = 0; e < 4; ++e) {
            float x = xs[e];
            int j = idx + e;
            if (x > av) { av = x; ai = j; }
            if (x > lm) { lsum = lsum * expf(lm - x) + 1.0f; lm = x; }
            else        { lsum += expf(x - lm); }
            atomicAdd(&hist[fkey(x) >> 21], 1u);
        }
    });

    redM[tid] = lm; redS[tid] = lsum; redAV[tid] = av; redAI[tid] = ai;
    __syncthreads();
    for (int s = NT / 2; s > 0; s >>= 1) {
        if (tid < s) {
            float m1 = redM[tid], s1 = redS[tid];
            float m2 = redM[tid + s], s2 = redS[tid + s];
            float m = fmaxf(m1, m2);
            float ns = 0.0f;
            if (m > -INFINITY) ns = s1 * expf(m1 - m) + s2 * expf(m2 - m);
            redM[tid] = m; redS[tid] = ns;
            float a1 = redAV[tid], a2 = redAV[tid + s];
            int   i1 = redAI[tid], i2 = redAI[tid + s];
            if (a2 > a1 || (a2 == a1 && i2 < i1)) { redAV[tid] = a2; redAI[tid] = i2; }
        }
        __syncthreads();
    }

    // radix level 0: top 11 bits (parallel suffix-scan selection)
    select_bin(hist, scal, scan, 21, tid);

    // ---- Passes 2,3 (L2): radix levels 1 (11 bits @10) and 2 (10 bits @0) ----
    for (int lvl = 1; lvl < 3; ++lvl) {
        const int sh = (lvl == 1) ? 10 : 0;
        const int bt = (lvl == 1) ? 11 : 10;
        const uint32_t hiShift = (uint32_t)(sh + bt);
        const uint32_t mask = (1u << bt) - 1u;
        for (int i = tid; i < NB; i += NT) hist[i] = 0u;   // full 2048; upper bins stay 0
        __syncthreads();
        const uint32_t prefix = scal[0];
        stream_row(row, V, smem, tid, [&](int idx, float4 v) {
            float xs[4] = {v.x, v.y, v.z, v.w};
            (void)idx;
#pragma unroll
            for (int e = 0; e < 4; ++e) {
                uint32_t k = fkey(xs[e]);
                if ((k >> hiShift) == (prefix >> hiShift))
                    atomicAdd(&hist[(k >> sh) & mask], 1u);
            }
        });
        __syncthreads();
        select_bin(hist, scal, scan, sh, tid);
    }

    // ---- Pass 4 (L2): gather top-N candidates ----
    if (tid == 0) { scal[3] = 0u; scal[4] = 0u; }
    __syncthreads();
    const uint32_t T = scal[0];   // exact key of the TOPN-th largest element
    stream_row(row, V, smem, tid, [&](int idx, float4 v) {
        float xs[4] = {v.x, v.y, v.z, v.w};
#pragma unroll
        for (int e = 0; e < 4; ++e) {
            uint32_t k = fkey(xs[e]);
            int j = idx + e;
            if (k > T) {                       // guaranteed < TOPN entries
                uint32_t p = atomicAdd(&scal[3], 1u);
                candV[p] = xs[e]; candI[p] = j;
            } else if (k == T) {
                uint32_t p = atomicAdd(&scal[4], 1u);
                if (p < CAND_CAP) { tieV[p] = xs[e]; tieI[p] = j; }
            }
        }
    });
    __syncthreads();

    // merge ties + sort (value desc, index asc) — tiny (<=50), serial on thread 0
    if (tid == 0) {
        int nG   = (int)scal[3];
        int nTs  = min((int)scal[4], CAND_CAP);
        int take = min((int)scal[1], nTs);
        for (int i = 0; i < take; ++i) { candV[nG + i] = tieV[i]; candI[nG + i] = tieI[i]; }
        int total = nG + take;
        for (int i = 1; i < total; ++i) {
            float v = candV[i]; int ix = candI[i];
            int j2 = i - 1;
            while (j2 >= 0 && (candV[j2] < v || (candV[j2] == v && candI[j2] > ix))) {
                candV[j2 + 1] = candV[j2]; candI[j2 + 1] = candI[j2]; --j2;
            }
            candV[j2 + 1] = v; candI[j2 + 1] = ix;
        }
        scal[5] = (uint32_t)total;
    }
    __syncthreads();

    // parallel gather of u at candidate indices (only ~50 values per row!)
    const int total = (int)scal[5];
    if (tid < total) uvals[tid] = u[(size_t)b * (size_t)V + (size_t)candI[tid]];
    __syncthreads();

    // ---- epilogue: sampling + outputs (scalar, thread 0) ----
    if (tid == 0) {
        const float M    = redM[0];
        const float logS = logf(redS[0]);
        const int greedy = redAI[0];
        int K = num_logprobs[0];
        if (K < 0) K = 0; if (K > total) K = total;
        const float traw = temperature[b];
        const bool useG  = traw < 1e-5f;
        const float invt = 1.0f / (useG ? 1.0f : traw);
        int k = top_k[b]; if (k < 1) k = 1; if (k > total) k = total;
        const float thr = 1.0f - top_p[b];

        // softmax over top-k scaled candidates (matches reference's masked softmax)
        const float s0 = candV[0] * invt;
        float ssum = 0.0f;
        for (int i = 0; i < k; ++i) ssum += expf(candV[i] * invt - s0);
        const float inv_ssum = 1.0f / ssum;

        // ascending cumulative mass == suffix sum in descending order;
        // keep i iff suffix > 1-p (largest candidate always kept); Gumbel-max.
        float best = -INFINITY; int bestIdx = 0; float suffix = 0.0f;
        for (int i = k - 1; i >= 0; --i) {
            float si = candV[i] * invt;
            suffix += expf(si - s0) * inv_ssum;
            if (i == 0 || suffix > thr) {
                float val = si - logf(-logf(uvals[i]));
                if (val >= best) { best = val; bestIdx = i; }
            }
        }
        const int   sampled = useG ? greedy : candI[bestIdx];
        const float sv      = useG ? M : candV[bestIdx];
        const float tok_lp  = sv - M - logS;
        int rank = 0;
        for (int i = 0; i < total; ++i) rank += (candV[i] > sv) ? 1 : 0;

        // outputs: [B] i32 sampled | [B,K+1] i32 indices | [B,K+1] f32 logprobs | [B] i32 ranks
        int*   oS = (int*)out;
        int*   oI = oS + B;
        float* oL = (float*)(oI + (size_t)B * (size_t)(K + 1));
        int*   oR = (int*)(oL + (size_t)B * (size_t)(K + 1));
        oS[b] = sampled;
        int*   rI = oI + (size_t)b * (size_t)(K + 1);
        float* rL = oL + (size_t)b * (size_t)(K + 1);
        rI[0] = sampled; rL[0] = tok_lp;
        for (int i = 0; i < K; ++i) { rI[1 + i] = candI[i]; rL[1 + i] = candV[i] - M - logS; }
        oR[b] = rank;
    }
}

extern "C" void kernel_launch(void* const* d_in, const int* in_sizes, int n_in,
                              void* d_out, int out_size, void* d_ws, size_t ws_size,
                              hipStream_t stream) {
    (void)n_in; (void)out_size; (void)d_ws; (void)ws_size;
    const float* logits       = (const float*)d_in[0];
    const float* temperature  = (const float*)d_in[1];
    const float* top_p        = (const float*)d_in[2];
    const float* u            = (const float*)d_in[3];
    const int*   top_k        = (const int*)d_in[4];
    const int*   num_logprobs = (const int*)d_in[5];
    const int B = in_sizes[1];
    const int V = in_sizes[0] / B;
    sampler_kernel<<<dim3(B), dim3(NT), SMEM_BYTES, stream>>>(
        logits, temperature, top_p, u, top_k, num_logprobs, (char*)d_out, B, V);
}